// AdaptiveTokenSampling_15573551415410
// MI455X (gfx1250) — compile-verified
//
#include <hip/hip_runtime.h>
#include <hip/hip_bf16.h>
#include <math.h>

#define EPS 1e-6f
#define B_  16
#define H_  12
#define NV  1025      // N
#define NM  1024      // N-1
#define D_  64
#define K_  256
#define KP1 257       // K+1
#define MASK_VALUE (-1.7014118346046923e38f)   // -float_max/2

typedef float v2f __attribute__((ext_vector_type(2)));
typedef float v8f __attribute__((ext_vector_type(8)));

// ---------------------------------------------------------------------------
// Kernel 1: value_norms[b,h,n] = ||value[b,h,1+n,:]|| via chained
// V_WMMA_F32_16X16X4_F32.  A (16x4) and B (4x16) register layouts are exact
// mirrors, so feeding the same registers as A and B yields D = A * A^T; the
// diagonal of the accumulated 16x16 Gram matrix is the squared norms.
// One wave handles one 16-row tile of one (b,h).
// ---------------------------------------------------------------------------
__global__ __launch_bounds__(256) void ats_norms_wmma(
    const float* __restrict__ value, float* __restrict__ norms) {
  const int wave = blockIdx.x * (blockDim.x >> 5) + (threadIdx.x >> 5);
  const int lane = threadIdx.x & 31;
  const int tile = wave & 63;          // 64 tiles of 16 rows per (b,h)
  const int bh   = wave >> 6;          // 0 .. B*H-1
  const int n0   = tile * 16;
  const int m    = lane & 15;          // A-matrix row this lane owns
  const int kh   = (lane >> 4) * 2;    // lanes 16..31 hold K=+2,+3

  const float* vp =
      value + ((size_t)bh * NV + (size_t)(1 + n0 + m)) * D_ + kh;

  v8f c = {};
#pragma unroll
  for (int d0 = 0; d0 < D_; d0 += 4) {
    v2f a = *(const v2f*)(vp + d0);    // 8B-aligned: d0,kh even
    c = __builtin_amdgcn_wmma_f32_16x16x4_f32(
        /*neg_a=*/false, a, /*neg_b=*/false, a,
        /*c_mod=*/(short)0, c, /*reuse_a=*/false, /*reuse_b=*/false);
  }

  // Extract diagonal D[m][m]:
  //   m in 0..7  -> lane m,      VGPR m
  //   m in 8..15 -> lane m+16,   VGPR m-8
  // Select chain (not runtime array index) so the accumulator stays in VGPRs.
  const int idx = (lane < 8) ? lane : (lane - 24);   // accumulator element
  float diag = c[0];
#pragma unroll
  for (int r = 1; r < 8; ++r) diag = (idx == r) ? c[r] : diag;

  if (lane < 8) {
    norms[(size_t)bh * NM + n0 + lane] = sqrtf(diag);
  } else if (lane >= 24) {
    norms[(size_t)bh * NM + n0 + (lane - 16)] = sqrtf(diag);
  }
}

// ---------------------------------------------------------------------------
// Kernel 2: scores -> normalize -> log -> mask => pseudo_logits[b, n]
// One block (1024 threads) per batch; only row 0 of attn per (b,h) is read.
// ---------------------------------------------------------------------------
__global__ __launch_bounds__(1024) void ats_pseudo_logits(
    const float* __restrict__ attn, const float* __restrict__ norms,
    const unsigned char* __restrict__ mask, float* __restrict__ plog) {
  const int b = blockIdx.x;
  const int n = threadIdx.x;   // 0..1023

  float s = 0.f;
#pragma unroll
  for (int h = 0; h < H_; ++h) {
    const size_t bh = (size_t)b * H_ + h;
    const float cls = attn[(bh * NV + 0) * NV + 1 + n];  // attn[b,h,0,1+n]
    s += cls * norms[bh * NM + n];
  }

  __shared__ float red[1024];
  red[n] = s;
  __syncthreads();
  for (int off = 512; off > 0; off >>= 1) {
    if (n < off) red[n] += red[n + off];
    __syncthreads();
  }
  const float total = red[0];

  const float pl = logf(s / (total + EPS) + EPS);
  const bool mk  = mask[b * NV + 1 + n] != 0;
  plog[b * NM + n] = mk ? pl : MASK_VALUE;
}

// ---------------------------------------------------------------------------
// Kernel 3: sampled[b,k] = argmax_n(plog[b,n] + gumbel[b,k,n]) + 1
// One wave per (b,k); first-index tie-breaking matches jnp.argmax.
// ---------------------------------------------------------------------------
__global__ __launch_bounds__(256) void ats_argmax(
    const float* __restrict__ plog, const float* __restrict__ gnoise,
    int* __restrict__ sampled) {
  const int wave = blockIdx.x * (blockDim.x >> 5) + (threadIdx.x >> 5);
  const int lane = threadIdx.x & 31;
  const int b = wave >> 8;       // K = 256
  const int k = wave & 255;

  const float* g  = gnoise + ((size_t)b * K_ + k) * NM;
  const float* pl = plog + (size_t)b * NM;

  float best = -INFINITY;
  int   bi   = 0;
  for (int n = lane; n < NM; n += 32) {
    const float gum = -logf(-logf(g[n] + EPS) + EPS);
    const float v = pl[n] + gum;
    if (v > best) { best = v; bi = n; }   // strict > keeps first max
  }
#pragma unroll
  for (int off = 16; off > 0; off >>= 1) {
    const float ob = __shfl_xor(best, off, 32);
    const int   oi = __shfl_xor(bi, off, 32);
    if (ob > best || (ob == best && oi < bi)) { best = ob; bi = oi; }
  }
  if (lane == 0) sampled[b * K_ + k] = bi + 1;
}

// ---------------------------------------------------------------------------
// Kernel 4: per-batch sort(256) -> zero duplicates -> re-sort -> ids / mask.
// One block (256 threads) per batch; bitonic sort in LDS.
// ---------------------------------------------------------------------------
__device__ __forceinline__ void bitonic256(int* s, int tid) {
  for (int k = 2; k <= 256; k <<= 1) {
    for (int j = k >> 1; j > 0; j >>= 1) {
      const int ixj = tid ^ j;
      if (ixj > tid) {
        const int a = s[tid], c = s[ixj];
        const bool asc = ((tid & k) == 0);
        if ((asc && a > c) || (!asc && a < c)) { s[tid] = c; s[ixj] = a; }
      }
      __syncthreads();
    }
  }
}

__global__ __launch_bounds__(256) void ats_sort_ids(
    const int* __restrict__ sampled, int* __restrict__ ids_int,
    float* __restrict__ out_mask, float* __restrict__ out_ids) {
  const int b = blockIdx.x;
  const int tid = threadIdx.x;
  __shared__ int s[256];

  s[tid] = sampled[b * K_ + tid];
  __syncthreads();
  bitonic256(s, tid);

  const int v = s[tid];
  const int isdup = (tid > 0 && s[tid] == s[tid - 1]) ? 1 : 0;
  __syncthreads();
  s[tid] = isdup ? 0 : v;
  __syncthreads();
  bitonic256(s, tid);

  const int u = s[tid];
  if (tid == 0) {
    ids_int[b * KP1] = 0;
    out_ids[b * KP1]  = 0.f;
    out_mask[b * KP1] = 1.f;   // padded True
  }
  ids_int[b * KP1 + 1 + tid]  = u;
  out_ids[b * KP1 + 1 + tid]  = (float)u;
  out_mask[b * KP1 + 1 + tid] = (u != 0) ? 1.f : 0.f;
}

// ---------------------------------------------------------------------------
// Kernel 5 (bandwidth-dominant): new_attn[b,h,j,:] = attn[b,h,ids[b,j],:]
// One block per gathered row (B*H*257 rows of 1025 floats); coalesced b32.
// ---------------------------------------------------------------------------
__global__ __launch_bounds__(256) void ats_gather(
    const float* __restrict__ attn, const int* __restrict__ ids_int,
    float* __restrict__ out_attn) {
  const int idx = blockIdx.x;        // b*H*257 + h*257 + j
  const int j   = idx % KP1;
  const int bh  = idx / KP1;
  const int b   = bh / H_;
  const int row = ids_int[b * KP1 + j];

  const float* src = attn + ((size_t)bh * NV + row) * NV;
  float* dst       = out_attn + (size_t)idx * NV;
  for (int n = threadIdx.x; n < NV; n += 256) dst[n] = src[n];
}

// ---------------------------------------------------------------------------
extern "C" void kernel_launch(void* const* d_in, const int* in_sizes, int n_in,
                              void* d_out, int out_size, void* d_ws, size_t ws_size,
                              hipStream_t stream) {
  const float*         attn   = (const float*)d_in[0];          // (B,H,N,N)
  const float*         value  = (const float*)d_in[1];          // (B,H,N,D)
  const float*         gnoise = (const float*)d_in[2];          // (B,K,N-1)
  const unsigned char* mask   = (const unsigned char*)d_in[3];  // (B,N) bool

  // Workspace layout
  float* norms   = (float*)d_ws;                     // B*H*NM  = 196608 f32
  float* plog    = norms + (size_t)B_ * H_ * NM;     // B*NM    = 16384  f32
  int*   sampled = (int*)(plog + (size_t)B_ * NM);   // B*K     = 4096   i32
  int*   ids_int = sampled + B_ * K_;                // B*KP1   = 4112   i32

  // Output layout (tuple flattened as float32)
  float* out_attn = (float*)d_out;                               // B*H*KP1*NV
  float* out_mask = out_attn + (size_t)B_ * H_ * KP1 * NV;       // B*KP1
  float* out_ids  = out_mask + (size_t)B_ * KP1;                 // B*KP1

  // 1) norms via WMMA Gram diagonals: B*H*64 tiles, 8 waves/block
  ats_norms_wmma<<<(B_ * H_ * 64) / 8, 256, 0, stream>>>(value, norms);

  // 2) pseudo-logits: one block per batch
  ats_pseudo_logits<<<B_, 1024, 0, stream>>>(attn, norms, mask, plog);

  // 3) gumbel argmax: B*K waves, 8 waves/block
  ats_argmax<<<(B_ * K_) / 8, 256, 0, stream>>>(plog, gnoise, sampled);

  // 4) sort + dedup + ids/mask
  ats_sort_ids<<<B_, 256, 0, stream>>>(sampled, ids_int, out_mask, out_ids);

  // 5) row gather (dominant ~405 MB of HBM traffic)
  ats_gather<<<B_ * H_ * KP1, 256, 0, stream>>>(attn, ids_int, out_attn);
}